// BaseCamera_1589137899573
// MI455X (gfx1250) — compile-verified
//
#include <hip/hip_runtime.h>
#include <math.h>

// ---------------- constants ----------------
#define WRES 4096
#define VRES 2048
#define TWO_PI_F 6.28318530717958647692f
#define PITCH_F 2e-6f
#define LAMBDA_F 5.32e-7f
#define D2_F 0.05f
#define WCROP 728
#define CROP_S 1685                       // 2048 - 364 + 1
#define KWAVE (TWO_PI_F / LAMBDA_F)       // 2*pi/lambda
#define INV_SD (1.0f / (4096.0f * PITCH_F))
#define INV_L2 ((1.0f / LAMBDA_F) * (1.0f / LAMBDA_F))

typedef __attribute__((ext_vector_type(2))) float v2f;
typedef __attribute__((ext_vector_type(8))) float v8f;

// D(16x16,f32) = A(16x4,f32) * B(4x16,f32) + C  — CDNA5 wave32 WMMA
static __device__ __forceinline__ v8f wmma4(v2f a, v2f b, v8f c) {
  return __builtin_amdgcn_wmma_f32_16x16x4_f32(false, a, false, b, (short)0, c,
                                               false, false);
}

// ---------------- 4096-pt Stockham radix-16 FFT in LDS (3 stages) -----------
// Input in buf0, result returned (lands in buf1). sgn = -1 fwd, +1 inv.
// Block = 256 threads = 8 waves; each wave does 2 groups of 16 butterflies
// per stage; each group = 16 x v_wmma_f32_16x16x4_f32 (complex 16x16 DFT).
static __device__ __forceinline__ float2* fft4096_lds(float2* buf0,
                                                      float2* buf1,
                                                      float sgn) {
  const int tid  = threadIdx.x;
  const int lane = tid & 31;
  const int wave = tid >> 5;
  const int Ml   = lane & 15;   // A-row / B-col / D-col index for this lane
  const int hi   = lane >> 4;   // lane-half select (K split, D-row split)

  // DFT16 matrix fragments: W[M][K] = exp(sgn*i*2pi*M*K/16)
  v2f war[4], wai[4], wan[4];
#pragma unroll
  for (int c = 0; c < 4; ++c) {
    const int kA = 4 * c + 2 * hi;
    const float a0 = sgn * (TWO_PI_F / 16.0f) * (float)(Ml * kA);
    const float a1 = sgn * (TWO_PI_F / 16.0f) * (float)(Ml * (kA + 1));
    war[c].x = __cosf(a0); war[c].y = __cosf(a1);
    wai[c].x = __sinf(a0); wai[c].y = __sinf(a1);
    wan[c].x = -wai[c].x;  wan[c].y = -wai[c].y;
  }

  float2* src = buf0;
  float2* dst = buf1;
#pragma unroll
  for (int stage = 0; stage < 3; ++stage) {
    const int s = (stage == 0) ? 1 : (stage == 1) ? 16 : 256;   // interleave
    const int m = (stage == 0) ? 256 : (stage == 1) ? 16 : 1;   // n/16
    const int n = (stage == 0) ? 4096 : (stage == 1) ? 256 : 16;
    __syncthreads();
#pragma unroll
    for (int gi = 0; gi < 2; ++gi) {
      const int bf = (wave * 2 + gi) * 16 + Ml;  // butterfly (column) index
      const int p  = bf / s;
      const int q  = bf - p * s;
      // Gather B fragments: C[r][col], r = K index of radix-16 butterfly
      v2f br[4], bi[4];
#pragma unroll
      for (int c = 0; c < 4; ++c) {
        const int r0 = 4 * c + 2 * hi;
        const float2 e0 = src[q + s * (p + m * r0)];
        const float2 e1 = src[q + s * (p + m * (r0 + 1))];
        br[c].x = e0.x; br[c].y = e1.x;
        bi[c].x = e0.y; bi[c].y = e1.y;
      }
      v8f dr = {0.f, 0.f, 0.f, 0.f, 0.f, 0.f, 0.f, 0.f};
      v8f di = {0.f, 0.f, 0.f, 0.f, 0.f, 0.f, 0.f, 0.f};
      // Dr = Wr*Cr - Wi*Ci ; Di = Wr*Ci + Wi*Cr  (K accumulated in 4 chunks)
#pragma unroll
      for (int c = 0; c < 4; ++c) {
        dr = wmma4(war[c], br[c], dr);
        di = wmma4(war[c], bi[c], di);
      }
#pragma unroll
      for (int c = 0; c < 4; ++c) {
        dr = wmma4(wan[c], bi[c], dr);
        di = wmma4(wai[c], br[c], di);
      }
      // Twiddle W_n^{p*k} and Stockham scatter
      const float tb = sgn * (TWO_PI_F / (float)n) * (float)p;
#pragma unroll
      for (int rr = 0; rr < 8; ++rr) {
        const int kk = rr + 8 * hi;        // D-matrix row held by this lane
        const float ang = tb * (float)kk;
        const float ct = __cosf(ang), st = __sinf(ang);
        const float xr = dr[rr], xi = di[rr];
        dst[q + s * (16 * p + kk)] = make_float2(xr * ct - xi * st,
                                                 xr * st + xi * ct);
      }
    }
    float2* t = src; src = dst; dst = t;
  }
  __syncthreads();
  return src;  // == buf1 after 3 swaps
}

// ---------------- kernel 1: spline tables (y, M) -----------------------------
__global__ void k_spline(const float* __restrict__ param, float* __restrict__ Y,
                         float* __restrict__ Msp) {
  __shared__ float cp[2046];
  __shared__ float dp[2046];
  const int t = threadIdx.x;
  for (int i = t; i < 2048; i += 256)
    Y[i] = (i < 1024) ? param[i >> 1] : 0.0f;   // repeat x2 then zero-pad
  __syncthreads();
  if (t == 0) {
    float cprev = 0.0f, dprev = 0.0f;
    for (int i = 0; i < 2046; ++i) {
      const float r = 6.0f * (Y[i + 2] - 2.0f * Y[i + 1] + Y[i]);
      const float den = 4.0f - cprev;
      cprev = 1.0f / den;
      dprev = (r - dprev) / den;
      cp[i] = cprev; dp[i] = dprev;
    }
    float x = 0.0f;
    Msp[2047] = 0.0f;
    for (int i = 2045; i >= 0; --i) {
      x = dp[i] - cp[i] * x;
      Msp[i + 1] = x;
    }
    Msp[0] = 0.0f;
  }
}

// ---------------- kernel 2: field synthesis ---------------------------------
__global__ __launch_bounds__(256)
void k_field(const float* __restrict__ Y, const float* __restrict__ Msp,
             float2* __restrict__ A) {
  const int idx = blockIdx.x * 256 + threadIdx.x;
  const int y = idx >> 12;
  const int x = idx & 4095;
  float2 v = make_float2(0.0f, 0.0f);
  if (y >= 1024 && y < 3072 && x >= 1024 && x < 3072) {
    const float dy = PITCH_F * (float)(y - 2048);
    const float dx = PITCH_F * (float)(x - 2048);
    const float r2 = dx * dx + dy * dy;
    // input_phase + lens_phase = k*r2*(1/(2*D1) - 1/(2*F)) = -10*k*r2
    const float ip = -10.0f * KWAVE * r2;
    // phase bias: mirrored quadrant spline eval
    const float ci = fabsf((float)(y - 1024) - 1023.5f);
    const float cj = fabsf((float)(x - 1024) - 1023.5f);
    const float r = sqrtf(ci * ci + cj * cj);
    int ind = (int)floorf(r);
    ind = min(max(ind, 0), 2046);
    const float tt = r - (float)ind;
    const float y0 = Y[ind], y1 = Y[ind + 1];
    const float m0 = Msp[ind], m1 = Msp[ind + 1];
    const float b = (y1 - y0) - (2.0f * m0 + m1) * (1.0f / 6.0f);
    const float bias =
        y0 + tt * (b + tt * (0.5f * m0 + tt * (m1 - m0) * (1.0f / 6.0f)));
    float ph = fmodf(ip + bias, TWO_PI_F);
    if (ph < 0.0f) ph += TWO_PI_F;
    v.x = __cosf(ph);
    v.y = __sinf(ph);
  }
  A[idx] = v;
}

// ---------------- kernel 3: row FFT pass (optionally fused H multiply) ------
__global__ __launch_bounds__(256)
void k_fft_rows(const float2* __restrict__ in, float2* __restrict__ out,
                float sgn, float scale, int applyH) {
  extern __shared__ float2 smem[];
  float2* buf0 = smem;
  float2* buf1 = smem + 4096;
  const int tid = threadIdx.x;
  const size_t base = (size_t)blockIdx.x * 4096u;
#pragma unroll
  for (int i = 0; i < 16; ++i) {
    const float2 v = in[base + tid + 256 * i];
    buf0[tid + 256 * i] = make_float2(v.x * scale, v.y * scale);
  }
  float2* res = fft4096_lds(buf0, buf1, sgn);
  if (applyH) {
    // fftshift folded into H index; |.|^2 later kills the (-1)^(x+y) term
    const int ru = ((blockIdx.x + 2048) & 4095);
    const float fxv = (float)(ru + 1 - 2048) * INV_SD;
    const float fx2 = INV_L2 - fxv * fxv;
#pragma unroll
    for (int i = 0; i < 16; ++i) {
      const int j = tid + 256 * i;
      const int ju = ((j + 2048) & 4095);
      const float fyv = (float)(ju + 1 - 2048) * INV_SD;
      const float w1 = sqrtf(fmaxf(fx2 - fyv * fyv, 0.0f));
      float hp = fmodf(TWO_PI_F * w1 * D2_F, TWO_PI_F);
      const float ch = __cosf(hp), sh = __sinf(hp);
      const float2 v = res[j];
      out[base + j] = make_float2(v.x * ch - v.y * sh, v.x * sh + v.y * ch);
    }
  } else {
#pragma unroll
    for (int i = 0; i < 16; ++i) out[base + tid + 256 * i] = res[tid + 256 * i];
  }
}

// ---------------- kernel 4: LDS-tiled transpose -----------------------------
__global__ void k_transpose(const float2* __restrict__ in,
                            float2* __restrict__ out) {
  __shared__ float2 tile[32][33];
  const int bx = blockIdx.x * 32, by = blockIdx.y * 32;
  const int tx = threadIdx.x, ty = threadIdx.y;  // 32 x 8
#pragma unroll
  for (int i = 0; i < 32; i += 8)
    tile[ty + i][tx] = in[(size_t)(by + ty + i) * 4096u + (bx + tx)];
  __syncthreads();
#pragma unroll
  for (int i = 0; i < 32; i += 8)
    out[(size_t)(bx + ty + i) * 4096u + (by + tx)] = tile[tx][ty + i];
}

// ---------------- kernel 5: final inverse row pass + |.|^2 + crop + sums ----
__global__ __launch_bounds__(256)
void k_fft_final(const float2* __restrict__ in, float* __restrict__ out,
                 float* __restrict__ partials) {
  extern __shared__ float2 smem[];
  __shared__ float red[256];
  float2* buf0 = smem;
  float2* buf1 = smem + 4096;
  const int tid = threadIdx.x;
  const size_t base = (size_t)(CROP_S + blockIdx.x) * 4096u;
  const float scale = 1.0f / 4096.0f;
#pragma unroll
  for (int i = 0; i < 16; ++i) {
    const float2 v = in[base + tid + 256 * i];
    buf0[tid + 256 * i] = make_float2(v.x * scale, v.y * scale);
  }
  float2* res = fft4096_lds(buf0, buf1, 1.0f);
  float local = 0.0f;
  for (int i = tid; i < WCROP; i += 256) {
    const float2 v = res[CROP_S + i];
    const float p = v.x * v.x + v.y * v.y;
    out[(size_t)blockIdx.x * WCROP + i] = p;
    local += p;
  }
  red[tid] = local;
  __syncthreads();
  for (int s2 = 128; s2 > 0; s2 >>= 1) {
    if (tid < s2) red[tid] += red[tid + s2];
    __syncthreads();
  }
  if (tid == 0) partials[blockIdx.x] = red[0];
}

// ---------------- kernels 6/7: deterministic reduce + normalize -------------
__global__ void k_reduce(const float* __restrict__ partials,
                         float* __restrict__ sumAcc) {
  __shared__ float red[256];
  const int t = threadIdx.x;
  float s = 0.0f;
  for (int i = t; i < WCROP; i += 256) s += partials[i];
  red[t] = s;
  __syncthreads();
  for (int k = 128; k > 0; k >>= 1) {
    if (t < k) red[t] += red[t + k];
    __syncthreads();
  }
  if (t == 0) sumAcc[0] = red[0];
}

__global__ void k_norm(float* __restrict__ out,
                       const float* __restrict__ sumAcc) {
  const int i = blockIdx.x * 256 + threadIdx.x;
  if (i < WCROP * WCROP) out[i] *= (1.0f / sumAcc[0]);
}

// ---------------- launcher --------------------------------------------------
extern "C" void kernel_launch(void* const* d_in, const int* in_sizes, int n_in,
                              void* d_out, int out_size, void* d_ws,
                              size_t ws_size, hipStream_t stream) {
  (void)in_sizes; (void)n_in; (void)out_size; (void)ws_size;
  const float* param = (const float*)d_in[0];
  float* out = (float*)d_out;
  float* ws = (float*)d_ws;
  // workspace layout (floats): Y[2048] | M[2048] | sum[1] pad | partials[728]
  // pad | A[4096^2 complex] | B[4096^2 complex]  (~268 MB total)
  float* Y        = ws;
  float* Msp      = ws + 2048;
  float* sumAcc   = ws + 4096;
  float* partials = ws + 4104;
  float2* A = (float2*)(ws + 4864);
  float2* B = A + (size_t)WRES * WRES;

  const unsigned lds = 2u * 4096u * sizeof(float2);  // 64 KB ping-pong

  k_spline<<<1, 256, 0, stream>>>(param, Y, Msp);
  k_field<<<(WRES * WRES) / 256, 256, 0, stream>>>(Y, Msp, A);
  // forward fft2: rows, transpose, rows (+H fused at store)
  k_fft_rows<<<WRES, 256, lds, stream>>>(A, B, -1.0f, 1.0f, 0);
  k_transpose<<<dim3(WRES / 32, WRES / 32), dim3(32, 8), 0, stream>>>(B, A);
  k_fft_rows<<<WRES, 256, lds, stream>>>(A, B, -1.0f, 1.0f, 1);
  // inverse fft2: rows (along ky), transpose, cropped rows (along kx)
  k_fft_rows<<<WRES, 256, lds, stream>>>(B, A, 1.0f, 1.0f / 4096.0f, 0);
  k_transpose<<<dim3(WRES / 32, WRES / 32), dim3(32, 8), 0, stream>>>(A, B);
  k_fft_final<<<WCROP, 256, lds, stream>>>(B, out, partials);
  k_reduce<<<1, 256, 0, stream>>>(partials, sumAcc);
  k_norm<<<(WCROP * WCROP + 255) / 256, 256, 0, stream>>>(out, sumAcc);
}